// FuseSliceCatSameInputModule_37469294690920
// MI455X (gfx1250) — compile-verified
//
#include <hip/hip_runtime.h>
#include <stdint.h>

// FuseSliceCatSameInput: out[r, off_j + k] = in[r, s_j + k]  (600 slices, L in {1..64})
// Pure data movement -> CDNA5 Tensor Data Mover (TDM) strided 2D tile copies
// global -> LDS -> global, double-buffered so store(t) overlaps load(t+1).
// Roofline: ~0.82 GB total traffic @ 23.3 TB/s HBM => ~35-70 us floor.

typedef unsigned int       u32;
typedef unsigned long long u64;
typedef unsigned int __attribute__((ext_vector_type(4))) v4u;
typedef int          __attribute__((ext_vector_type(4))) v4i;
typedef int          __attribute__((ext_vector_type(8))) v8i;

#define SC_ROWS     8192
#define SC_COLS     8192
#define SC_OUTCOLS  12500
#define SC_ROWTILE  128   // rows per TDM descriptor
#define SC_NT       4     // row-tiles per wave (pipelined); 512 rows per wave
#define SC_MAXLEN   64    // max slice length (elements)

// ---------------------------------------------------------------------------
// Kernel A: exclusive prefix scan of slice lengths -> output column offsets.
// ---------------------------------------------------------------------------
__global__ void sc_build_offsets(const int* __restrict__ slices,
                                 int* __restrict__ offs, int nslices) {
  if (blockIdx.x == 0 && threadIdx.x == 0) {
    int acc = 0;
    for (int j = 0; j < nslices; ++j) {
      offs[j] = acc;
      acc += slices[2 * j + 1] - slices[2 * j];
    }
    offs[nslices] = acc;
  }
}

// ---------------------------------------------------------------------------
// TDM descriptor (D#) for a 2D tile: `rows` x `len` f32, global row pitch
// `row_stride` elements, rows packed in LDS. Packing per CDNA5 ISA §8.3/§8.4;
// groups 2/3 zero (tile_dim2 == 0 marks dims >= 2 unused).
// ---------------------------------------------------------------------------
__device__ __forceinline__ void sc_make_desc(u64 gaddr, u32 lds_byte_addr,
                                             u32 len, u32 rows, u32 row_stride,
                                             v4u& g0, v8i& g1, v4i& g2, v4i& g3) {
  // group0: [1:0]=count(1 valid), [63:32]=lds_addr, [120:64]=global_addr,
  //         [127:126]=type(2)
  g0[0] = 1u;
  g0[1] = lds_byte_addr;
  g0[2] = (u32)(gaddr & 0xFFFFFFFFull);
  g0[3] = (u32)((gaddr >> 32) & 0x01FFFFFFull) | (2u << 30);

  // group1: [17:16]=data_size(2 => 4B), [79:48]=tensor_dim0=len,
  //         [111:80]=tensor_dim1=rows, [127:112]=tile_dim0=len,
  //         [143:128]=tile_dim1=rows, [159:144]=tile_dim2=0,
  //         [207:160]=tensor_dim0_stride, [255:208]=tensor_dim1_stride=0
  g1[0] = (int)(2u << 16);
  g1[1] = (int)((len & 0xFFFFu) << 16);
  g1[2] = (int)(((len >> 16) & 0xFFFFu) | ((rows & 0xFFFFu) << 16));
  g1[3] = (int)(((rows >> 16) & 0xFFFFu) | ((len & 0xFFFFu) << 16));
  g1[4] = (int)(rows & 0xFFFFu);
  g1[5] = (int)row_stride;
  g1[6] = 0;
  g1[7] = 0;

  g2[0] = 0; g2[1] = 0; g2[2] = 0; g2[3] = 0;
  g3[0] = 0; g3[1] = 0; g3[2] = 0; g3[3] = 0;
}

__device__ __forceinline__ void sc_tdm_load(u64 gaddr, u32 lds, u32 len) {
  v4u g0; v8i g1; v4i g2, g3;
  sc_make_desc(gaddr, lds, len, (u32)SC_ROWTILE, (u32)SC_COLS, g0, g1, g2, g3);
#if defined(__clang_major__) && (__clang_major__ >= 23)
  v8i gz; gz[0]=0; gz[1]=0; gz[2]=0; gz[3]=0; gz[4]=0; gz[5]=0; gz[6]=0; gz[7]=0;
  __builtin_amdgcn_tensor_load_to_lds(g0, g1, g2, g3, gz, 0);
#else
  __builtin_amdgcn_tensor_load_to_lds(g0, g1, g2, g3, 0);
#endif
}

__device__ __forceinline__ void sc_tdm_store(u64 gaddr, u32 lds, u32 len) {
  v4u g0; v8i g1; v4i g2, g3;
  sc_make_desc(gaddr, lds, len, (u32)SC_ROWTILE, (u32)SC_OUTCOLS, g0, g1, g2, g3);
#if defined(__clang_major__) && (__clang_major__ >= 23)
  v8i gz; gz[0]=0; gz[1]=0; gz[2]=0; gz[3]=0; gz[4]=0; gz[5]=0; gz[6]=0; gz[7]=0;
  __builtin_amdgcn_tensor_store_from_lds(g0, g1, g2, g3, gz, 0);
#else
  __builtin_amdgcn_tensor_store_from_lds(g0, g1, g2, g3, 0);
#endif
}

// ---------------------------------------------------------------------------
// Kernel B: one wave per (slice, group of SC_NT row-tiles). TDM ignores EXEC,
// so the wave issues one DMA per tensor instruction. Double-buffered pipeline:
//   L(0); for t: { wait_tensorcnt 0; S(t); L(t+1); }
// The wait-0 guarantees (per-wave in-order completion) that L(t) finished
// (so S(t) may read buf t&1) and that S(t-1) finished (so L(t+1) may
// overwrite buf (t+1)&1). S(t) and L(t+1) then run concurrently.
// ---------------------------------------------------------------------------
__global__ __launch_bounds__(32, 1)
void sc_tdm_slice_cat(const float* __restrict__ in,
                      const int* __restrict__ slices,
                      const int* __restrict__ offs,
                      float* __restrict__ out) {
  __shared__ float tile[2][SC_ROWTILE * SC_MAXLEN];  // 2 x 32 KB

  const int j    = blockIdx.x;                        // slice index
  const int row0 = blockIdx.y * (SC_ROWTILE * SC_NT); // first row of this wave
  const int s    = slices[2 * j];
  const int e    = slices[2 * j + 1];
  const u32 L    = (u32)(e - s);
  const int off  = offs[j];

  const u32 lds0 = (u32)(uintptr_t)&tile[0][0];  // low 32b of flat LDS addr
  const u32 lds1 = (u32)(uintptr_t)&tile[1][0];

  u64 gin  = (u64)(uintptr_t)in  + 4ull * ((u64)row0 * SC_COLS    + (u64)s);
  u64 gout = (u64)(uintptr_t)out + 4ull * ((u64)row0 * SC_OUTCOLS + (u64)off);
  const u64 gin_step  = 4ull * (u64)SC_ROWTILE * SC_COLS;
  const u64 gout_step = 4ull * (u64)SC_ROWTILE * SC_OUTCOLS;

  // Prologue: fill buffer 0.
  sc_tdm_load(gin, lds0, L);

#pragma unroll
  for (int t = 0; t < SC_NT; ++t) {
    const u32 lds_cur  = (t & 1) ? lds1 : lds0;
    const u32 lds_next = (t & 1) ? lds0 : lds1;

    __builtin_amdgcn_s_wait_tensorcnt(0);  // L(t) done; S(t-1) done

    sc_tdm_store(gout + (u64)t * gout_step, lds_cur, L);
    if (t + 1 < SC_NT)
      sc_tdm_load(gin + (u64)(t + 1) * gin_step, lds_next, L);
  }

  __builtin_amdgcn_s_wait_tensorcnt(0);    // final store complete
}

// ---------------------------------------------------------------------------
extern "C" void kernel_launch(void* const* d_in, const int* in_sizes, int n_in,
                              void* d_out, int out_size, void* d_ws, size_t ws_size,
                              hipStream_t stream) {
  const float* in     = (const float*)d_in[0];
  const int*   slices = (const int*)d_in[1];   // [nslices, 2] int32
  float*       out    = (float*)d_out;
  int*         offs   = (int*)d_ws;            // nslices+1 ints of scratch

  const int nslices = in_sizes[1] / 2;         // 600

  sc_build_offsets<<<1, 32, 0, stream>>>(slices, offs, nslices);

  dim3 grid((unsigned)nslices, SC_ROWS / (SC_ROWTILE * SC_NT));  // 600 x 16 waves
  sc_tdm_slice_cat<<<grid, 32, 0, stream>>>(in, slices, offs, out);

  (void)n_in; (void)out_size; (void)ws_size;
}